// Edge_Embedder_75900662055248
// MI455X (gfx1250) — compile-verified
//
#include <hip/hip_runtime.h>

typedef float v2f __attribute__((ext_vector_type(2)));
typedef float v8f __attribute__((ext_vector_type(8)));

#define N_NODES 100000
#define N_EDGES 1600000
#define IN_D 50
#define HID_D 200
#define OUT_D 100

// padded weight shapes (pair-major: [K/2][Npad][2])
#define K1P 52
#define N1P 208
#define K2P 200
#define N2P 112
#define W1P_TOTAL ((K1P / 2) * N1P * 2)   // 10816
#define W2P_TOTAL ((K2P / 2) * N2P * 2)   // 22400

// ---------------- init: deg=1 (self loop), out=0 ----------------
__global__ void k_init(unsigned* __restrict__ deg, float* __restrict__ out) {
    int i = blockIdx.x * blockDim.x + threadIdx.x;
    if (i < N_NODES) deg[i] = 1u;
    if (i < OUT_D)   out[i] = 0.0f;
}

// ---------------- build zero-padded, pair-major weight copies ----------------
__global__ void k_padW(const float* __restrict__ W1, const float* __restrict__ W2,
                       float* __restrict__ W1p, float* __restrict__ W2p) {
    int idx = blockIdx.x * blockDim.x + threadIdx.x;
    if (idx < W1P_TOTAL) {
        int j = idx & 1, t = idx >> 1;
        int n = t % N1P, p = t / N1P;
        int k = 2 * p + j;
        W1p[idx] = (k < IN_D && n < HID_D) ? W1[k * HID_D + n] : 0.0f;
    }
    if (idx < W2P_TOTAL) {
        int j = idx & 1, t = idx >> 1;
        int n = t % N2P, p = t / N2P;
        int k = 2 * p + j;
        W2p[idx] = (k < HID_D && n < OUT_D) ? W2[k * OUT_D + n] : 0.0f;
    }
}

// ---------------- degree: atomic count of incoming edges ----------------
__global__ void k_degree(const int* __restrict__ dst, unsigned* __restrict__ deg) {
    int e = blockIdx.x * blockDim.x + threadIdx.x;
    if (e < N_EDGES) atomicAdd(&deg[dst[e]], 1u);
}

__global__ void k_dinv(const unsigned* __restrict__ deg, float* __restrict__ dinv) {
    int i = blockIdx.x * blockDim.x + threadIdx.x;
    if (i < N_NODES) dinv[i] = rsqrtf((float)deg[i]);
}

// ---------------- GEMM1: g1 = dinv * (x @ W1); acc1 = g1 (self-loop init) ----
// 13 waves/block; each wave owns one 16-wide N tile of a 16-row M tile.
__global__ __launch_bounds__(416) void k_gemm1(const float* __restrict__ x,
                                               const float* __restrict__ W1p,
                                               const float* __restrict__ dinv,
                                               float* __restrict__ g1,
                                               float* __restrict__ acc1) {
    __shared__ float As[16][53];        // K padded 50 -> 52, stride 53 (bank-friendly)
    const int M0 = blockIdx.x * 16;
    for (int idx = threadIdx.x; idx < 16 * K1P; idx += blockDim.x) {
        int r = idx / K1P, k = idx % K1P;
        As[r][k] = (k < IN_D) ? x[(M0 + r) * IN_D + k] : 0.0f;
    }
    __syncthreads();

    const int wave = threadIdx.x >> 5;
    const int lane = threadIdx.x & 31;
    const int half = lane >> 4;         // 0: K pair {0,1}, 1: K pair {2,3}
    const int l    = lane & 15;
    const int n    = wave * 16 + l;     // < 208 always

    // pair-major: element (k, n) lives at W1p[(k/2)*N1P*2 + n*2 + (k&1)]
    const float* wp = W1p + 2 * n + half * (2 * N1P);

    v8f c = {};
    for (int k0 = 0; k0 < K1P; k0 += 4) {
        const int kb = k0 + 2 * half;
        v2f a, b;
        a.x = As[l][kb];
        a.y = As[l][kb + 1];
        b = *reinterpret_cast<const v2f*>(wp + k0 * N1P);   // one b64 load
        c = __builtin_amdgcn_wmma_f32_16x16x4_f32(false, a, false, b,
                                                  (short)0, c, false, false);
    }
    if (n < HID_D) {
#pragma unroll
        for (int r = 0; r < 8; ++r) {
            const int m = M0 + r + 8 * half;
            const float v = c[r] * dinv[m];
            g1[m * HID_D + n]   = v;
            acc1[m * HID_D + n] = v;
        }
    }
}

// ---------------- GEMM2: g2 = dinv * (h @ W2); acc2 = g2 ----------------
// K=200 (divisible by 4); 7 waves/block.
__global__ __launch_bounds__(224) void k_gemm2(const float* __restrict__ h,
                                               const float* __restrict__ W2p,
                                               const float* __restrict__ dinv,
                                               float* __restrict__ g2,
                                               float* __restrict__ acc2) {
    __shared__ float As[16][201];
    const int M0 = blockIdx.x * 16;
    for (int idx = threadIdx.x; idx < 16 * K2P; idx += blockDim.x) {
        int r = idx / K2P, k = idx % K2P;
        As[r][k] = h[(M0 + r) * K2P + k];
    }
    __syncthreads();

    const int wave = threadIdx.x >> 5;
    const int lane = threadIdx.x & 31;
    const int half = lane >> 4;
    const int l    = lane & 15;
    const int n    = wave * 16 + l;     // < 112 always

    const float* wp = W2p + 2 * n + half * (2 * N2P);

    v8f c = {};
    for (int k0 = 0; k0 < K2P; k0 += 4) {
        const int kb = k0 + 2 * half;
        v2f a, b;
        a.x = As[l][kb];
        a.y = As[l][kb + 1];
        b = *reinterpret_cast<const v2f*>(wp + k0 * N2P);
        c = __builtin_amdgcn_wmma_f32_16x16x4_f32(false, a, false, b,
                                                  (short)0, c, false, false);
    }
    if (n < OUT_D) {
#pragma unroll
        for (int r = 0; r < 8; ++r) {
            const int m = M0 + r + 8 * half;
            const float v = c[r] * dinv[m];
            g2[m * OUT_D + n]   = v;
            acc2[m * OUT_D + n] = v;
        }
    }
}

// ------- edge scatter: acc[dst] += g[src], 4 features per thread ----------
template <int F>
__global__ void k_scatter(const int* __restrict__ src, const int* __restrict__ dst,
                          const float* __restrict__ g, float* __restrict__ acc) {
    constexpr int F4 = F / 4;
    const long long idx = (long long)blockIdx.x * blockDim.x + threadIdx.x;
    const long long total = (long long)N_EDGES * F4;
    if (idx < total) {
        const int e = (int)(idx / F4);
        const int q = (int)(idx % F4);
        const int s = src[e];
        const int d = dst[e];
        const float4 v = reinterpret_cast<const float4*>(g)[(long long)s * F4 + q];
        float* a = acc + (long long)d * F + 4 * q;
        atomicAdd(a + 0, v.x);
        atomicAdd(a + 1, v.y);
        atomicAdd(a + 2, v.z);
        atomicAdd(a + 3, v.w);
    }
}

// ---------------- layer-1 epilogue: h = relu(dinv*acc1 + b1) ----------------
__global__ void k_post1(const float* __restrict__ acc1, const float* __restrict__ dinv,
                        const float* __restrict__ b1, float* __restrict__ h) {
    const long long idx = (long long)blockIdx.x * blockDim.x + threadIdx.x;
    if (idx < (long long)N_NODES * HID_D) {
        const int i = (int)(idx / HID_D);
        const int f = (int)(idx % HID_D);
        const float v = dinv[i] * acc1[idx] + b1[f];
        h[idx] = v > 0.0f ? v : 0.0f;
    }
}

// ------- layer-2 epilogue + column mean: out[f] = mean_i relu(...) --------
__global__ __launch_bounds__(128) void k_mean(const float* __restrict__ acc2,
                                              const float* __restrict__ dinv,
                                              const float* __restrict__ b2,
                                              float* __restrict__ out) {
    const int f = threadIdx.x;
    const int r0 = blockIdx.x * 250;
    const int r1 = r0 + 250;
    if (f < OUT_D) {
        const float bias = b2[f];
        float s = 0.0f;
        for (int r = r0; r < r1; ++r) {
            const float v = dinv[r] * acc2[(long long)r * OUT_D + f] + bias;
            s += (v > 0.0f) ? v : 0.0f;
        }
        atomicAdd(&out[f], s * (1.0f / (float)N_NODES));
    }
}

extern "C" void kernel_launch(void* const* d_in, const int* in_sizes, int n_in,
                              void* d_out, int out_size, void* d_ws, size_t ws_size,
                              hipStream_t stream) {
    const float* x  = (const float*)d_in[0];
    const int*   ei = (const int*)d_in[1];     // [2, E]
    const float* W1 = (const float*)d_in[2];
    const float* b1 = (const float*)d_in[3];
    const float* W2 = (const float*)d_in[4];
    const float* b2 = (const float*)d_in[5];
    float* out = (float*)d_out;

    const int* src = ei;
    const int* dst = ei + N_EDGES;

    // workspace layout (floats): deg | dinv | W1p | W2p | g1 | acc1 | g2 | acc2
    float* ws     = (float*)d_ws;
    unsigned* deg = (unsigned*)ws;
    float* dinv   = ws + N_NODES;
    float* W1p    = dinv + N_NODES;
    float* W2p    = W1p + W1P_TOTAL;
    float* g1     = W2p + W2P_TOTAL;
    float* acc1   = g1 + (size_t)N_NODES * HID_D;
    float* g2     = acc1 + (size_t)N_NODES * HID_D;
    float* acc2   = g2 + (size_t)N_NODES * OUT_D;

    k_init<<<(N_NODES + 255) / 256, 256, 0, stream>>>(deg, out);
    k_padW<<<(W2P_TOTAL + 255) / 256, 256, 0, stream>>>(W1, W2, W1p, W2p);
    k_degree<<<(N_EDGES + 255) / 256, 256, 0, stream>>>(dst, deg);
    k_dinv<<<(N_NODES + 255) / 256, 256, 0, stream>>>(deg, dinv);

    // layer 1
    k_gemm1<<<N_NODES / 16, 416, 0, stream>>>(x, W1p, dinv, g1, acc1);
    {
        const long long total = (long long)N_EDGES * (HID_D / 4);
        k_scatter<HID_D><<<(unsigned)((total + 255) / 256), 256, 0, stream>>>(src, dst, g1, acc1);
    }
    k_post1<<<(N_NODES * HID_D + 255) / 256, 256, 0, stream>>>(acc1, dinv, b1, g1);

    // layer 2
    k_gemm2<<<N_NODES / 16, 224, 0, stream>>>(g1, W2p, dinv, g2, acc2);
    {
        const long long total = (long long)N_EDGES * (OUT_D / 4);
        k_scatter<OUT_D><<<(unsigned)((total + 255) / 256), 256, 0, stream>>>(src, dst, g2, acc2);
    }
    k_mean<<<N_NODES / 250, 128, 0, stream>>>(acc2, dinv, b2, out);
}